// Net_87557203296258
// MI455X (gfx1250) — compile-verified
//
#include <hip/hip_runtime.h>
#include <hip/hip_bf16.h>
#include <math.h>

typedef __attribute__((ext_vector_type(2))) float v2f;
typedef __attribute__((ext_vector_type(8))) float v8f;

#define FIN   128
#define HIDN  128
#define CLS   40
#define CPAD  48

// ---------------------------------------------------------------- degree prep
__global__ void deg_init_kernel(float* deg, int n) {
    int i = blockIdx.x * blockDim.x + threadIdx.x;
    if (i < n) deg[i] = 1.0f;                    // self-loop contribution
}

__global__ void deg_count_kernel(const int* __restrict__ dst, float* deg, int e) {
    int i = blockIdx.x * blockDim.x + threadIdx.x;
    if (i < e) atomicAdd(&deg[dst[i]], 1.0f);
}

__global__ void deg_rsqrt_kernel(float* deg, int n) {
    int i = blockIdx.x * blockDim.x + threadIdx.x;
    if (i < n) deg[i] = rsqrtf(deg[i]);          // in place -> inv_sqrt
}

// ---------------------------------------------------------------- GEMM1: h1 = x @ W1  (N x 128 @ 128 x 128), fp32 WMMA
__global__ __launch_bounds__(256) void gemm1_kernel(const float* __restrict__ x,
                                                    const float* __restrict__ W1,
                                                    float* __restrict__ h1) {
    __shared__ float ldsW[FIN * HIDN];   // 64 KB
    __shared__ float ldsX[16 * FIN];     //  8 KB
    const int tid = threadIdx.x;
    const int rowBase = blockIdx.x * 16;

    for (int i = tid; i < FIN * HIDN / 4; i += 256)
        ((float4*)ldsW)[i] = ((const float4*)W1)[i];
    for (int i = tid; i < 16 * FIN / 4; i += 256)
        ((float4*)ldsX)[i] = ((const float4*)(x + (size_t)rowBase * FIN))[i];
    __syncthreads();

    const int wave  = tid >> 5;
    const int lane  = tid & 31;
    const int m     = lane & 15;          // A row / D column index
    const int khalf = (lane >> 4) * 2;    // 32-bit A: hi half-wave holds K+2,K+3
    const int colBase = wave * 16;        // 8 waves -> 8 column tiles of 16

    v8f c = {};
#pragma unroll
    for (int k = 0; k < FIN; k += 4) {
        v2f a = *(const v2f*)&ldsX[m * FIN + k + khalf];
        v2f b;
        b.x = ldsW[(k + khalf) * HIDN + colBase + m];
        b.y = ldsW[(k + khalf + 1) * HIDN + colBase + m];
        c = __builtin_amdgcn_wmma_f32_16x16x4_f32(false, a, false, b,
                                                  (short)0, c, false, false);
    }

    const int mrow = rowBase + 8 * (lane >> 4);   // C/D: VGPR v -> M = v + 8*(lane>=16)
#pragma unroll
    for (int v = 0; v < 8; ++v)
        h1[(size_t)(mrow + v) * HIDN + colBase + m] = c[v];
}

// ---------------------------------------------------------------- layer-1 self term + bias init
__global__ void init_agg1_kernel(const float* __restrict__ h1,
                                 const float* __restrict__ invs,
                                 const float* __restrict__ b1,
                                 float* __restrict__ agg1, int n) {
    int t = blockIdx.x * blockDim.x + threadIdx.x;
    if (t >= n * 32) return;
    int node = t >> 5;
    int q    = (t & 31) * 4;
    float is = invs[node];
    float s  = is * is;                   // self-loop norm = 1/deg
    const float4 h = *(const float4*)(h1 + (size_t)node * HIDN + q);
    float4 o;
    o.x = b1[q + 0] + h.x * s;
    o.y = b1[q + 1] + h.y * s;
    o.z = b1[q + 2] + h.z * s;
    o.w = b1[q + 3] + h.w * s;
    *(float4*)(agg1 + (size_t)node * HIDN + q) = o;
}

// ---------------------------------------------------------------- layer-1 edge scatter (wave per edge)
__global__ __launch_bounds__(256) void scatter1_kernel(const int* __restrict__ src,
                                                       const int* __restrict__ dst,
                                                       const float* __restrict__ invs,
                                                       const float* __restrict__ h1,
                                                       float* __restrict__ agg1, int e) {
    int w = (blockIdx.x * blockDim.x + threadIdx.x) >> 5;
    int lane = threadIdx.x & 31;
    if (w >= e) return;
    int s = src[w], d = dst[w];
    float nrm = invs[s] * invs[d];
    const float4 v = *(const float4*)(h1 + (size_t)s * HIDN + lane * 4);
    float* out = agg1 + (size_t)d * HIDN + lane * 4;
    atomicAdd(out + 0, v.x * nrm);
    atomicAdd(out + 1, v.y * nrm);
    atomicAdd(out + 2, v.z * nrm);
    atomicAdd(out + 3, v.w * nrm);
}

// ---------------------------------------------------------------- GEMM2: h2 = relu(agg1) @ W2pad  (N x 128 @ 128 x 48)
__global__ __launch_bounds__(256) void gemm2_kernel(const float* __restrict__ agg1,
                                                    const float* __restrict__ W2,
                                                    float* __restrict__ h2, int ntiles) {
    __shared__ float ldsW[HIDN * CPAD];   // 24 KB, zero padded cols 40..47
    const int tid = threadIdx.x;
    for (int i = tid; i < HIDN * CPAD; i += 256) {
        int r = i / CPAD, col = i % CPAD;
        ldsW[i] = (col < CLS) ? W2[r * CLS + col] : 0.0f;
    }
    __syncthreads();

    const int wave = tid >> 5;
    const int lane = tid & 31;
    const int flat = blockIdx.x * 8 + wave;
    if (flat >= ntiles) return;
    const int tm = flat / 3, tn = flat % 3;
    const int rowBase = tm * 16, colBase = tn * 16;
    const int m = lane & 15;
    const int khalf = (lane >> 4) * 2;
    const float* arow = agg1 + (size_t)(rowBase + m) * HIDN;

    v8f c = {};
#pragma unroll
    for (int k = 0; k < HIDN; k += 4) {
        v2f av = *(const v2f*)&arow[k + khalf];
        v2f a, b;
        a.x = fmaxf(av.x, 0.0f);          // fused ReLU
        a.y = fmaxf(av.y, 0.0f);
        b.x = ldsW[(k + khalf) * CPAD + colBase + m];
        b.y = ldsW[(k + khalf + 1) * CPAD + colBase + m];
        c = __builtin_amdgcn_wmma_f32_16x16x4_f32(false, a, false, b,
                                                  (short)0, c, false, false);
    }

    const int mrow = rowBase + 8 * (lane >> 4);
#pragma unroll
    for (int v = 0; v < 8; ++v)
        h2[(size_t)(mrow + v) * CPAD + colBase + m] = c[v];
}

// ---------------------------------------------------------------- layer-2 self term + bias init
__global__ void init_agg2_kernel(const float* __restrict__ h2,
                                 const float* __restrict__ invs,
                                 const float* __restrict__ b2,
                                 float* __restrict__ agg2, int n) {
    int t = blockIdx.x * blockDim.x + threadIdx.x;
    if (t >= n * CLS) return;
    int node = t / CLS;
    int j    = t % CLS;
    float is = invs[node];
    agg2[t] = b2[j] + h2[(size_t)node * CPAD + j] * is * is;
}

// ---------------------------------------------------------------- layer-2 edge scatter (wave per edge, 40 cols)
__global__ __launch_bounds__(256) void scatter2_kernel(const int* __restrict__ src,
                                                       const int* __restrict__ dst,
                                                       const float* __restrict__ invs,
                                                       const float* __restrict__ h2,
                                                       float* __restrict__ agg2, int e) {
    int w = (blockIdx.x * blockDim.x + threadIdx.x) >> 5;
    int lane = threadIdx.x & 31;
    if (w >= e) return;
    int s = src[w], d = dst[w];
    float nrm = invs[s] * invs[d];
    float v0 = h2[(size_t)s * CPAD + lane];
    atomicAdd(agg2 + (size_t)d * CLS + lane, v0 * nrm);
    if (lane < CLS - 32) {
        float v1 = h2[(size_t)s * CPAD + 32 + lane];
        atomicAdd(agg2 + (size_t)d * CLS + 32 + lane, v1 * nrm);
    }
}

// ---------------------------------------------------------------- wave-per-row log_softmax over 40 classes
__global__ __launch_bounds__(256) void logsoftmax_kernel(const float* __restrict__ agg2,
                                                         float* __restrict__ out, int n) {
    int row  = (blockIdx.x * blockDim.x + threadIdx.x) >> 5;
    int lane = threadIdx.x & 31;
    if (row >= n) return;
    float v0 = agg2[(size_t)row * CLS + lane];
    float v1 = (lane < 8) ? agg2[(size_t)row * CLS + 32 + lane] : -INFINITY;
    float mx = fmaxf(v0, v1);
#pragma unroll
    for (int off = 16; off >= 1; off >>= 1)
        mx = fmaxf(mx, __shfl_xor(mx, off, 32));
    float s = expf(v0 - mx) + ((lane < 8) ? expf(v1 - mx) : 0.0f);
#pragma unroll
    for (int off = 16; off >= 1; off >>= 1)
        s += __shfl_xor(s, off, 32);
    float lg = logf(s);
    out[(size_t)row * CLS + lane] = v0 - mx - lg;
    if (lane < 8)
        out[(size_t)row * CLS + 32 + lane] = v1 - mx - lg;
}

// ---------------------------------------------------------------- launch
extern "C" void kernel_launch(void* const* d_in, const int* in_sizes, int n_in,
                              void* d_out, int out_size, void* d_ws, size_t ws_size,
                              hipStream_t stream) {
    const float* x  = (const float*)d_in[0];
    const int* eidx = (const int*)d_in[1];
    const float* W1 = (const float*)d_in[2];
    const float* b1 = (const float*)d_in[3];
    const float* W2 = (const float*)d_in[4];
    const float* b2 = (const float*)d_in[5];
    float* out = (float*)d_out;

    const int n = in_sizes[0] / FIN;      // 100000
    const int e = in_sizes[1] / 2;        // 1600000
    const int* src = eidx;
    const int* dst = eidx + e;

    // workspace layout (floats):
    //   [0, n)                 : deg -> inv_sqrt
    //   bufA = align(n)        : h1 (n*128); later reused: h2 (n*48) + agg2 (n*40)
    //   bufB = bufA + n*128    : agg1 (n*128)
    float* ws   = (float*)d_ws;
    size_t offA = ((size_t)n + 63) & ~(size_t)63;
    float* invs = ws;
    float* bufA = ws + offA;
    float* agg1 = bufA + (size_t)n * HIDN;
    float* h1   = bufA;
    float* h2   = bufA;
    float* agg2 = bufA + (size_t)n * CPAD;

    const int TPB = 256;

    // degrees -> inv_sqrt
    deg_init_kernel<<<(n + TPB - 1) / TPB, TPB, 0, stream>>>(invs, n);
    deg_count_kernel<<<(e + TPB - 1) / TPB, TPB, 0, stream>>>(dst, invs, e);
    deg_rsqrt_kernel<<<(n + TPB - 1) / TPB, TPB, 0, stream>>>(invs, n);

    // layer 1
    gemm1_kernel<<<n / 16, TPB, 0, stream>>>(x, W1, h1);
    init_agg1_kernel<<<((size_t)n * 32 + TPB - 1) / TPB, TPB, 0, stream>>>(h1, invs, b1, agg1, n);
    scatter1_kernel<<<(e + 7) / 8, TPB, 0, stream>>>(src, dst, invs, h1, agg1, e);

    // layer 2 (ReLU fused into GEMM2 A-load)
    const int ntiles = (n / 16) * 3;
    gemm2_kernel<<<(ntiles + 7) / 8, TPB, 0, stream>>>(agg1, W2, h2, ntiles);
    init_agg2_kernel<<<((size_t)n * CLS + TPB - 1) / TPB, TPB, 0, stream>>>(h2, invs, b2, agg2, n);
    scatter2_kernel<<<(e + 7) / 8, TPB, 0, stream>>>(src, dst, invs, h2, agg2, e);

    // log_softmax -> output
    logsoftmax_kernel<<<(n + 7) / 8, TPB, 0, stream>>>(agg2, out, n);
}